// GCN_39247411151276
// MI455X (gfx1250) — compile-verified
//
#include <hip/hip_runtime.h>
#include <stdint.h>

#define THREADS 256
#define D 64

typedef unsigned int u32x4 __attribute__((ext_vector_type(4)));
typedef int i32x8 __attribute__((ext_vector_type(8)));
typedef int i32x4 __attribute__((ext_vector_type(4)));

// ---------------------------------------------------------------------------
// JAX threefry2x32 (20 rounds, key injection every 4) for the deterministic
// "graph dropout" index draw with key(42) = (0, 42).
// ---------------------------------------------------------------------------
__device__ __forceinline__ void threefry2x32(uint32_t k0, uint32_t k1,
                                             uint32_t x0, uint32_t x1,
                                             uint32_t& o0, uint32_t& o1) {
  const uint32_t ks[3] = {k0, k1, k0 ^ k1 ^ 0x1BD11BDAu};
  const uint32_t rot[8] = {13u, 15u, 26u, 6u, 17u, 29u, 16u, 24u};
  x0 += ks[0];
  x1 += ks[1];
#pragma unroll
  for (int r = 0; r < 5; ++r) {
#pragma unroll
    for (int i = 0; i < 4; ++i) {
      const uint32_t rr = rot[(r & 1) * 4 + i];
      x0 += x1;
      x1 = (x1 << rr) | (x1 >> (32u - rr));
      x1 ^= x0;
    }
    x0 += ks[(r + 1) % 3];
    x1 += ks[(r + 2) % 3] + (uint32_t)(r + 1);
  }
  o0 = x0;
  o1 = x1;
}

// ---------------------------------------------------------------------------
// Tensor Data Mover: 1-D tile load (global -> LDS). tensor_dim0 = avail gives
// hardware OOB clamping (reads past the tensor return 0) -> free tail handling.
// ---------------------------------------------------------------------------
#if __has_builtin(__builtin_amdgcn_tensor_load_to_lds)
#define HAVE_TDM 1
__device__ __forceinline__ void tdm_load_tile_1d(const void* gsrc, uint32_t lds_off,
                                                 uint32_t tile_elems, uint32_t avail) {
  const uint64_t ga = (uint64_t)(uintptr_t)gsrc;
  const uint32_t td0 = avail;      // tensor_dim0 (elements, OOB -> zero-fill)
  const uint32_t td1 = 1u;         // single line
  const uint64_t st0 = (uint64_t)td0;
  // D# group0: count=1 | lds_addr | global_addr[56:0] | type=2
  u32x4 g0 = {1u, lds_off, (uint32_t)ga,
              ((uint32_t)((ga >> 32) & 0x01FFFFFFu)) | 0x80000000u};
  // D# group1: data_size=4B; tensor_dim0/1; tile_dim0 = tile_elems; stride0
  i32x8 g1 = {(int)0x00020000u,
              (int)((td0 & 0xFFFFu) << 16),
              (int)((td0 >> 16) | ((td1 & 0xFFFFu) << 16)),
              (int)((td1 >> 16) | ((tile_elems & 0xFFFFu) << 16)),
              0,
              (int)(uint32_t)(st0 & 0xFFFFFFFFu),
              (int)(uint32_t)((st0 >> 32) & 0xFFFFu),
              0};
  i32x4 gz = {0, 0, 0, 0};
#if __clang_major__ >= 23
  i32x8 gz8 = {0, 0, 0, 0, 0, 0, 0, 0};
  __builtin_amdgcn_tensor_load_to_lds(g0, g1, gz, gz, gz8, 0);
#else
  __builtin_amdgcn_tensor_load_to_lds(g0, g1, gz, gz, 0);
#endif
}
#endif

__device__ __forceinline__ void async_row_b64(uint32_t lds_off, const float* gaddr) {
  asm volatile("global_load_async_to_lds_b64 %0, %1, off" ::"v"(lds_off), "v"(gaddr)
               : "memory");
}

// ---------------------------------------------------------------------------
// Setup kernels
// ---------------------------------------------------------------------------
__global__ __launch_bounds__(THREADS) void init_deg_kernel(float* __restrict__ dout,
                                                           float* __restrict__ din,
                                                           int N) {
  const int i = (int)(blockIdx.x * blockDim.x + threadIdx.x);
  if (i < N) {
    dout[i] = 1.0f;  // self-loop contributes 1 to out-degree
    din[i] = 1.0f;   // and 1 to in-degree (clamp(deg,1) automatic)
  }
}

__global__ __launch_bounds__(THREADS) void deg_kernel(const int* __restrict__ src,
                                                      const int* __restrict__ dst,
                                                      float* __restrict__ dout,
                                                      float* __restrict__ din, int E) {
  const int e = (int)(blockIdx.x * blockDim.x + threadIdx.x);
  if (e < E) {
    atomicAdd(&dout[src[e]], 1.0f);
    atomicAdd(&din[dst[e]], 1.0f);
  }
}

__global__ __launch_bounds__(THREADS) void rsqrt_kernel(float* __restrict__ dout,
                                                        float* __restrict__ din, int N) {
  const int i = (int)(blockIdx.x * blockDim.x + threadIdx.x);
  if (i < N) {
    dout[i] = rsqrtf(dout[i]);
    din[i] = rsqrtf(din[i]);
  }
}

__global__ __launch_bounds__(THREADS) void copy_wf_kernel(float* __restrict__ wf,
                                                          const float* __restrict__ w,
                                                          int E) {
  const int e = (int)(blockIdx.x * blockDim.x + threadIdx.x);
  if (e < E) wf[e] = w[e];
}

// Zero w_f at threefry-drawn indices (idx >= E hits zero self-loop weights).
__global__ __launch_bounds__(THREADS) void drop_kernel(float* __restrict__ wf, int n,
                                                       int half, int E, unsigned Ef) {
  const int t = (int)(blockIdx.x * blockDim.x + threadIdx.x);
  if (t >= half) return;
  const uint32_t x0 = (uint32_t)t;
  const uint32_t x1 = (t + half < n) ? (uint32_t)(t + half) : 0u;  // odd-pad = 0
  uint32_t o0, o1;
  threefry2x32(0u, 42u, x0, x1, o0, o1);
  const uint32_t i0 = o0 % Ef;
  if ((int)i0 < E) wf[i0] = 0.0f;
  if (half + t < n) {
    const uint32_t i1 = o1 % Ef;
    if ((int)i1 < E) wf[i1] = 0.0f;
  }
}

// ---------------------------------------------------------------------------
// float4 streaming kernels (i indexes float4; node = (4*i)/64 = i>>4)
// ---------------------------------------------------------------------------
__global__ __launch_bounds__(THREADS) void zero4_kernel(float4* __restrict__ p,
                                                        unsigned total4) {
  const unsigned i = blockIdx.x * blockDim.x + threadIdx.x;
  if (i < total4) p[i] = make_float4(0.f, 0.f, 0.f, 0.f);
}

// hs = h * dinv_out[node]
__global__ __launch_bounds__(THREADS) void scale4_kernel(float4* __restrict__ hs,
                                                         const float4* __restrict__ h,
                                                         const float* __restrict__ dinv,
                                                         unsigned total4) {
  const unsigned i = blockIdx.x * blockDim.x + threadIdx.x;
  if (i < total4) {
    __builtin_prefetch(h + i + 8192, 0, 0);
    const float s = dinv[i >> 4];
    float4 v = h[i];
    v.x *= s; v.y *= s; v.z *= s; v.w *= s;
    hs[i] = v;
  }
}

// e1 = acc * dinv_in[node] + e0
__global__ __launch_bounds__(THREADS) void fin4_kernel(float4* __restrict__ e1,
                                                       const float4* __restrict__ acc,
                                                       const float* __restrict__ dinv_in,
                                                       const float4* __restrict__ e0,
                                                       unsigned total4) {
  const unsigned i = blockIdx.x * blockDim.x + threadIdx.x;
  if (i < total4) {
    __builtin_prefetch(acc + i + 8192, 0, 0);
    const float s = dinv_in[i >> 4];
    const float4 a = acc[i];
    const float4 z = e0[i];
    e1[i] = make_float4(fmaf(a.x, s, z.x), fmaf(a.y, s, z.y), fmaf(a.z, s, z.z),
                        fmaf(a.w, s, z.w));
  }
}

// out = (e0 + 2*e1 + acc2*dinv_in[node]) / 3
__global__ __launch_bounds__(THREADS) void out4_kernel(float4* __restrict__ out,
                                                       const float4* __restrict__ e0,
                                                       const float4* __restrict__ e1,
                                                       const float4* __restrict__ acc,
                                                       const float* __restrict__ dinv_in,
                                                       unsigned total4) {
  const unsigned i = blockIdx.x * blockDim.x + threadIdx.x;
  if (i < total4) {
    __builtin_prefetch(acc + i + 8192, 0, 0);
    const float s = dinv_in[i >> 4];
    const float4 a = acc[i];
    const float4 z = e0[i];
    const float4 o = e1[i];
    const float k = 1.0f / 3.0f;
    out[i] = make_float4((z.x + 2.f * o.x + a.x * s) * k, (z.y + 2.f * o.y + a.y * s) * k,
                         (z.z + 2.f * o.z + a.z * s) * k, (z.w + 2.f * o.w + a.w * s) * k);
  }
}

// ---------------------------------------------------------------------------
// Edge scatter.
//  - TDM (tensor_load_to_lds) bulk-stages the 256-edge (src,dst,w) tile into
//    LDS from wave 0; OOB clamp (tensor_dim0 = E-base) zero-fills the tail,
//    so w=0 edges are skipped naturally.
//  - Each wave walks its 32 edges; the random 256B feature-row gather is
//    software-pipelined through an LDS double buffer with per-lane
//    global_load_async_to_lds_b64 + s_wait_asynccnt 1 (async loads complete
//    in order).
//  - Messages accumulate via f32 global atomics (avg in-degree ~12).
// ---------------------------------------------------------------------------
__global__ __launch_bounds__(THREADS) void scatter_kernel(
    const float* __restrict__ hs, const float* __restrict__ wf,
    const int* __restrict__ src, const int* __restrict__ dst,
    float* __restrict__ acc, int E) {
  __shared__ int sSrc[THREADS];
  __shared__ int sDst[THREADS];
  __shared__ float sW[THREADS];
  __shared__ float sRow[8][2][D];  // 8 waves x double buffer x 64 floats = 4KB

  const int t = (int)threadIdx.x;
  const int base = (int)blockIdx.x * THREADS;

#ifdef HAVE_TDM
  if (t < 32) {  // wave 0 issues the three tile descriptors
    const uint32_t avail = (uint32_t)(E - base);
    tdm_load_tile_1d(src + base, (uint32_t)(uintptr_t)(&sSrc[0]), THREADS, avail);
    tdm_load_tile_1d(dst + base, (uint32_t)(uintptr_t)(&sDst[0]), THREADS, avail);
    tdm_load_tile_1d(wf + base, (uint32_t)(uintptr_t)(&sW[0]), THREADS, avail);
#if __has_builtin(__builtin_amdgcn_s_wait_tensorcnt)
    __builtin_amdgcn_s_wait_tensorcnt(0);
#else
    asm volatile("s_wait_tensorcnt 0x0" ::: "memory");
#endif
  }
#else
  {
    const int ge = base + t;
    if (ge < E) {
      asm volatile("global_load_async_to_lds_b32 %0, %1, off" ::"v"(
                       (uint32_t)(uintptr_t)(&sSrc[t])),
                   "v"(src + ge)
                   : "memory");
      asm volatile("global_load_async_to_lds_b32 %0, %1, off" ::"v"(
                       (uint32_t)(uintptr_t)(&sDst[t])),
                   "v"(dst + ge)
                   : "memory");
      asm volatile("global_load_async_to_lds_b32 %0, %1, off" ::"v"(
                       (uint32_t)(uintptr_t)(&sW[t])),
                   "v"(wf + ge)
                   : "memory");
    } else {
      sSrc[t] = 0;
      sDst[t] = 0;
      sW[t] = 0.0f;
    }
    asm volatile("s_wait_asynccnt 0x0" ::: "memory");
  }
#endif
  __syncthreads();

  const int wave = t >> 5;  // wave32: 8 waves per block
  const int lane = t & 31;
  const int ebase = wave << 5;
  const uint32_t rb0 = (uint32_t)(uintptr_t)(&sRow[wave][0][lane * 2]);
  const uint32_t rb1 = (uint32_t)(uintptr_t)(&sRow[wave][1][lane * 2]);

  // Prolog: stream row of edge 0 into buffer 0.
  {
    const int s0 = sSrc[ebase];
    async_row_b64(rb0, hs + (size_t)s0 * D + lane * 2);
  }

#pragma unroll 1
  for (int j = 0; j < 32; ++j) {
    if (j < 31) {  // stream next row into the other buffer, then gate on j
      const int sn = sSrc[ebase + j + 1];
      async_row_b64((j & 1) ? rb0 : rb1, hs + (size_t)sn * D + lane * 2);
      asm volatile("s_wait_asynccnt 0x1" ::: "memory");
    } else {
      asm volatile("s_wait_asynccnt 0x0" ::: "memory");
    }
    const float wt = sW[ebase + j];  // wave-uniform (LDS broadcast)
    if (wt != 0.0f) {                // dropped edge / tail: no message
      const int d = sDst[ebase + j];
      const float* rp = (j & 1) ? &sRow[wave][1][lane * 2] : &sRow[wave][0][lane * 2];
      const float2 v = *(const float2*)rp;
      float* ap = acc + (size_t)d * D + lane * 2;
      atomicAdd(ap + 0, wt * v.x);
      atomicAdd(ap + 1, wt * v.y);
    }
  }
}

// ---------------------------------------------------------------------------
// Host launcher
// ---------------------------------------------------------------------------
extern "C" void kernel_launch(void* const* d_in, const int* in_sizes, int n_in,
                              void* d_out, int out_size, void* d_ws, size_t ws_size,
                              hipStream_t stream) {
  const float* w = (const float*)d_in[0];
  const float* feat = (const float*)d_in[1];  // e0, [N, 64]
  const int* src = (const int*)d_in[2];
  const int* dst = (const int*)d_in[3];
  float* out = (float*)d_out;

  const int E = in_sizes[0];
  const int N = in_sizes[1] / D;
  const unsigned total = (unsigned)N * D;
  const unsigned total4 = total >> 2;

  // Workspace (floats): dinv_out[N] | dinv_in[N] | wf[E] | hs[N*64] |
  // acc[N*64] | e1[N*64]   (~82 MB; all 16B-aligned for these sizes)
  float* ws = (float*)d_ws;
  float* dinv_out = ws;
  float* dinv_in = ws + N;
  float* wf = ws + 2 * (size_t)N;
  float* hs = wf + E;
  float* acc = hs + (size_t)N * D;
  float* e1 = acc + (size_t)N * D;

  const int Ef = E + N;                      // edges after add_self_loop
  const int drop = (int)(0.3 * (double)Ef);  // int(0.3 * E_f)
  const int half = (drop + 1) / 2;

  const int gN = (N + THREADS - 1) / THREADS;
  const int gE = (E + THREADS - 1) / THREADS;
  const int gT4 = (int)((total4 + THREADS - 1) / THREADS);
  const int gH = (half + THREADS - 1) / THREADS;

  // Degrees (self-loops folded into init) -> dinv = deg^-1/2
  init_deg_kernel<<<gN, THREADS, 0, stream>>>(dinv_out, dinv_in, N);
  deg_kernel<<<gE, THREADS, 0, stream>>>(src, dst, dinv_out, dinv_in, E);
  rsqrt_kernel<<<gN, THREADS, 0, stream>>>(dinv_out, dinv_in, N);

  // Edge weights + threefry dropout
  copy_wf_kernel<<<gE, THREADS, 0, stream>>>(wf, w, E);
  drop_kernel<<<gH, THREADS, 0, stream>>>(wf, drop, half, E, (unsigned)Ef);

  // conv #1: e1 = D_in^-1/2 A_w D_out^-1/2 e0 + e0
  zero4_kernel<<<gT4, THREADS, 0, stream>>>((float4*)acc, total4);
  scale4_kernel<<<gT4, THREADS, 0, stream>>>((float4*)hs, (const float4*)feat, dinv_out,
                                             total4);
  scatter_kernel<<<gE, THREADS, 0, stream>>>(hs, wf, src, dst, acc, E);
  fin4_kernel<<<gT4, THREADS, 0, stream>>>((float4*)e1, (const float4*)acc, dinv_in,
                                           (const float4*)feat, total4);

  // conv #2 + fused mean: out = (e0 + 2*e1 + acc2*dinv_in) / 3
  zero4_kernel<<<gT4, THREADS, 0, stream>>>((float4*)acc, total4);
  scale4_kernel<<<gT4, THREADS, 0, stream>>>((float4*)hs, (const float4*)e1, dinv_out,
                                             total4);
  scatter_kernel<<<gE, THREADS, 0, stream>>>(hs, wf, src, dst, acc, E);
  out4_kernel<<<gT4, THREADS, 0, stream>>>((float4*)out, (const float4*)feat,
                                           (const float4*)e1, (const float4*)acc,
                                           dinv_in, total4);
}